// TFiLM_2173253452614
// MI455X (gfx1250) — compile-verified
//
#include <hip/hip_runtime.h>

#define BLOCK_T 128
#define CCH     32      // channels == LSTM hidden
#define NP      2       // n_params
#define KX      36      // C+P padded to multiple of 4 (zeros in cols 34,35)
#define KC      68      // concatenated K: [w_ih(34) | pad(2) | w_hh(32)]
#define GATES   128     // 4*H

typedef __attribute__((ext_vector_type(2))) float v2f;
typedef __attribute__((ext_vector_type(8))) float v8f;

__device__ __forceinline__ float sigf(float v) {
    return 1.0f / (1.0f + __expf(-v));
}

// ---------------------------------------------------------------------------
// Kernel 0: one-time weight prep. wcat[col][0..33] = w_ih[col], [34..35] = 0,
// [36..67] = w_hh[col]; bias[col] = b_ih[col] + b_hh[col]. Removes all
// K-guards from the WMMA inner loop.
// ---------------------------------------------------------------------------
__global__ __launch_bounds__(256)
void tfilm_prep_kernel(const float* __restrict__ w_ih, const float* __restrict__ w_hh,
                       const float* __restrict__ b_ih, const float* __restrict__ b_hh,
                       float* __restrict__ wcat, float* __restrict__ bias) {
    int idx = blockIdx.x * blockDim.x + threadIdx.x;
    if (idx < GATES) bias[idx] = b_ih[idx] + b_hh[idx];
    for (int i = idx; i < GATES * KC; i += gridDim.x * blockDim.x) {
        int col = i / KC;
        int k   = i % KC;
        float v;
        if (k < 34)      v = w_ih[col * 34 + k];
        else if (k < 36) v = 0.0f;
        else             v = w_hh[col * CCH + (k - 36)];
        wcat[i] = v;
    }
}

// ---------------------------------------------------------------------------
// Kernel 1: blockwise maxpool (kernel=128) + transpose + p concat + zero pad.
// One wave per (b, ch, s): 32 lanes x float4 = 128 contiguous floats, wave
// max-reduce via shuffles. Writes xds[t=b][s][KX] so WMMA A-operands are
// contiguous per row (cols 34,35 zeroed to match wcat's pad).
// ---------------------------------------------------------------------------
__global__ __launch_bounds__(256)
void tfilm_pool_kernel(const float* __restrict__ x, const float* __restrict__ p,
                       float* __restrict__ xds, int B, int C, int T, int nsteps) {
    int wid  = threadIdx.x >> 5;
    int lane = threadIdx.x & 31;
    long widx  = (long)blockIdx.x * 8 + wid;
    long total = (long)B * C * nsteps;
    if (widx >= total) return;
    int s  = (int)(widx % nsteps);
    int ch = (int)((widx / nsteps) % C);
    int b  = (int)(widx / ((long)nsteps * C));

    const float4* src =
        reinterpret_cast<const float4*>(x + (long)(b * C + ch) * T + (long)s * BLOCK_T);
    float4 v = src[lane];
    float m = fmaxf(fmaxf(v.x, v.y), fmaxf(v.z, v.w));
    #pragma unroll
    for (int off = 16; off > 0; off >>= 1)
        m = fmaxf(m, __shfl_xor(m, off, 32));

    if (lane == 0) {
        long row = ((long)b * nsteps + s) * KX;
        xds[row + ch] = m;
        if (ch == 0) {                      // once per (b,s): params + zero pad
            xds[row + 32] = p[b * NP + 0];
            xds[row + 33] = p[b * NP + 1];
            xds[row + 34] = 0.0f;
            xds[row + 35] = 0.0f;
        }
    }
}

// ---------------------------------------------------------------------------
// Kernel 2: LSTM over the batch axis (seq len = B = 8) via
// V_WMMA_F32_16X16X4_F32. One wave per 16-row M-tile (grid = nsteps/16).
// Per step: gates(16x128) = [x(36)|h(32)] @ wcat^T + bias, K = 68.
// B operands are preloaded into distinct registers (one load clause) before
// the WMMA chain, and accumulation runs as two interleaved chains to cut the
// serial WMMA RAW depth — the recurrence is latency-bound, not BW-bound.
//
// WMMA f32 16x16x4 layouts (ISA 7.12.2):
//   A(16x4): lane holds M=lane&15; vgpr0/1 = K = 2*half / 2*half+1
//   B(4x16): lane holds N=lane&15; vgpr0/1 = K rows 2*half / 2*half+1
//   D:       vgpr r -> row r+8*half, col lane&15
// ---------------------------------------------------------------------------
__global__ __launch_bounds__(32)
void tfilm_lstm_kernel(const float* __restrict__ xds, const float* __restrict__ wcat,
                       const float* __restrict__ bias, float* __restrict__ h_out,
                       int B, int nsteps) {
    __shared__ float gates[16 * GATES];   // 8 KB
    __shared__ float htile[16 * CCH];     // 2 KB

    const int lane = threadIdx.x;         // 0..31, full wave (EXEC all ones)
    const int half = lane >> 4;
    const int nl   = lane & 15;
    const int mbase = blockIdx.x * 16;

    for (int i = lane; i < 16 * CCH; i += 32) htile[i] = 0.0f;
    __syncthreads();

    float c0[8], c1[8];
    #pragma unroll
    for (int r = 0; r < 8; ++r) { c0[r] = 0.0f; c1[r] = 0.0f; }

    for (int t = 0; t < B; ++t) {
        // ---- A operands for this step (shared across all 8 N-tiles) ----
        const float* xa = xds + ((long)t * nsteps + mbase) * KX + nl * KX + 2 * half;
        const float* ha = htile + nl * CCH + 2 * half;
        float a0[17], a1[17];
        #pragma unroll
        for (int q = 0; q < 9; ++q)  { a0[q] = xa[4 * q];       a1[q] = xa[4 * q + 1]; }
        #pragma unroll
        for (int q = 9; q < 17; ++q) { a0[q] = ha[4 * (q - 9)]; a1[q] = ha[4 * (q - 9) + 1]; }

        // ---- gates = [x|h] @ wcat^T + bias, tiled 16x16 over N ----
        for (int nt = 0; nt < 8; ++nt) {
            const int col = nt * 16 + nl;                 // gate column 0..127
            const float* wb = wcat + col * KC + 2 * half;

            // Preload all B operands: independent of the WMMAs -> one clause.
            float b0[17], b1[17];
            #pragma unroll
            for (int q = 0; q < 17; ++q) { b0[q] = wb[4 * q]; b1[q] = wb[4 * q + 1]; }

            // Two interleaved accumulation chains (halved RAW depth).
            v8f acce = {0.f, 0.f, 0.f, 0.f, 0.f, 0.f, 0.f, 0.f};
            v8f acco = {0.f, 0.f, 0.f, 0.f, 0.f, 0.f, 0.f, 0.f};
            #pragma unroll
            for (int q = 0; q < 17; ++q) {                // K = 68, unguarded
                v2f a  = {a0[q], a1[q]};
                v2f bm = {b0[q], b1[q]};
                if (q & 1)
                    acco = __builtin_amdgcn_wmma_f32_16x16x4_f32(
                               false, a, false, bm, (short)0, acco, false, false);
                else
                    acce = __builtin_amdgcn_wmma_f32_16x16x4_f32(
                               false, a, false, bm, (short)0, acce, false, false);
            }
            float bv = bias[col];
            #pragma unroll
            for (int r = 0; r < 8; ++r)                   // D: row r+8*half, col nl
                gates[(r + 8 * half) * GATES + col] = acce[r] + acco[r] + bv;
        }
        __syncthreads();

        // ---- cell update: i,f,g,o -> c,h (each lane: 8 rows x 2 channels) ----
        #pragma unroll
        for (int r = 0; r < 8; ++r) {
            const int row = r + 8 * half;
            {
                const int ch = nl;                         // channels 0..15
                float gi = gates[row * GATES + ch];
                float gf = gates[row * GATES + 32 + ch];
                float gg = gates[row * GATES + 64 + ch];
                float go = gates[row * GATES + 96 + ch];
                float c  = sigf(gf) * c0[r] + sigf(gi) * tanhf(gg);
                c0[r] = c;
                float h = sigf(go) * tanhf(c);
                htile[row * CCH + ch] = h;
                h_out[((long)t * nsteps + mbase + row) * CCH + ch] = h;
            }
            {
                const int ch = nl + 16;                    // channels 16..31
                float gi = gates[row * GATES + ch];
                float gf = gates[row * GATES + 32 + ch];
                float gg = gates[row * GATES + 64 + ch];
                float go = gates[row * GATES + 96 + ch];
                float c  = sigf(gf) * c1[r] + sigf(gi) * tanhf(gg);
                c1[r] = c;
                float h = sigf(go) * tanhf(c);
                htile[row * CCH + ch] = h;
                h_out[((long)t * nsteps + mbase + row) * CCH + ch] = h;
            }
        }
        __syncthreads();
    }
}

// ---------------------------------------------------------------------------
// Kernel 3: FiLM multiply, full-bandwidth streaming. One wave per (b,ch,s)
// block: 128 contiguous floats scaled by h_out[b][s][ch].
// ---------------------------------------------------------------------------
__global__ __launch_bounds__(256)
void tfilm_mul_kernel(const float* __restrict__ x, const float* __restrict__ h_out,
                      float* __restrict__ out, int B, int C, int T, int nsteps) {
    int wid  = threadIdx.x >> 5;
    int lane = threadIdx.x & 31;
    long widx  = (long)blockIdx.x * 8 + wid;
    long total = (long)B * C * nsteps;
    if (widx >= total) return;
    int s  = (int)(widx % nsteps);
    int ch = (int)((widx / nsteps) % C);
    int b  = (int)(widx / ((long)nsteps * C));

    float scale = h_out[((long)b * nsteps + s) * CCH + ch];
    long off = (long)(b * C + ch) * T + (long)s * BLOCK_T;
    const float4* src = reinterpret_cast<const float4*>(x + off);
    float4*       dst = reinterpret_cast<float4*>(out + off);
    float4 v = src[lane];
    v.x *= scale; v.y *= scale; v.z *= scale; v.w *= scale;
    dst[lane] = v;
}

extern "C" void kernel_launch(void* const* d_in, const int* in_sizes, int n_in,
                              void* d_out, int out_size, void* d_ws, size_t ws_size,
                              hipStream_t stream) {
    const float* x    = (const float*)d_in[0];
    const float* p    = (const float*)d_in[1];
    const float* w_ih = (const float*)d_in[2];
    const float* w_hh = (const float*)d_in[3];
    const float* b_ih = (const float*)d_in[4];
    const float* b_hh = (const float*)d_in[5];
    float* out = (float*)d_out;

    const int C = CCH;
    const int B = in_sizes[1] / NP;                         // 8
    const int T = (int)((long)in_sizes[0] / ((long)B * C)); // 262144
    const int nsteps = T / BLOCK_T;                         // 2048

    float* xds   = (float*)d_ws;                            // (B, nsteps, KX)
    float* h_out = xds + (long)B * nsteps * KX;             // (B, nsteps, C)
    float* wcat  = h_out + (long)B * nsteps * CCH;          // (128, 68)
    float* bias  = wcat + (long)GATES * KC;                 // (128,)

    long nblk = (long)B * C * nsteps;                       // one wave each
    dim3 gstream((unsigned)((nblk + 7) / 8));

    tfilm_prep_kernel<<<34, 256, 0, stream>>>(w_ih, w_hh, b_ih, b_hh, wcat, bias);
    tfilm_pool_kernel<<<gstream, 256, 0, stream>>>(x, p, xds, B, C, T, nsteps);
    tfilm_lstm_kernel<<<nsteps / 16, 32, 0, stream>>>(xds, wcat, bias, h_out, B, nsteps);
    tfilm_mul_kernel<<<gstream, 256, 0, stream>>>(x, h_out, out, B, C, T, nsteps);
}